// Encoder_GNN_81071802679527
// MI455X (gfx1250) — compile-verified
//
#include <hip/hip_runtime.h>

typedef __attribute__((ext_vector_type(2))) float v2f;
typedef __attribute__((ext_vector_type(8))) float v8f;
typedef int i4v __attribute__((vector_size(16)));  // int4, matches builtin param

#define N_NODES 100000
#define DFEAT   128

// gfx1250 async global->LDS load path (ASYNCcnt-tracked), with fallback.
#if defined(__gfx1250__) && __has_builtin(__builtin_amdgcn_global_load_async_to_lds_b128)
#define ASYNC_LDS 1
#else
#define ASYNC_LDS 0
#endif

__device__ __forceinline__ void wait_async_all() {
#if defined(__gfx1250__)
#if __has_builtin(__builtin_amdgcn_s_wait_asynccnt)
  __builtin_amdgcn_s_wait_asynccnt(0);
#else
  asm volatile("s_wait_asynccnt 0" ::: "memory");
#endif
#endif
}

// ---------------------------------------------------------------------------
// Zero out the aggregation buffer (float4 stores).
// ---------------------------------------------------------------------------
__global__ void zero_kernel(float* __restrict__ p, long long n4) {
  long long i = (long long)blockIdx.x * blockDim.x + threadIdx.x;
  if (i < n4) ((float4*)p)[i] = make_float4(0.f, 0.f, 0.f, 0.f);
}

// ---------------------------------------------------------------------------
// Edge scatter: one wave32 per edge. Each lane handles 4 of the 128 features
// (b128 load from the source row, 4x global_atomic_add_f32 into dest row).
// sigmoid(edge_weight) recomputed inline (cheap VALU vs. 6.4MB of traffic).
// Node features (51MB) are L2-resident on MI455X (192MB L2), so both the
// gather and the atomics resolve at L2's atomic units.
// ---------------------------------------------------------------------------
__global__ __launch_bounds__(256) void scatter_kernel(
    const float* __restrict__ feat, const long long* __restrict__ eidx,
    const float* __restrict__ ew, float* __restrict__ aggr, int E) {
  int edge = blockIdx.x * (blockDim.x >> 5) + (threadIdx.x >> 5);
  if (edge >= E) return;
  int lane = threadIdx.x & 31;
  long long src = eidx[edge];
  long long dst = eidx[(long long)E + edge];
  float w = ew[edge];
  w = 1.0f / (1.0f + __expf(-w));  // sigmoid
  float4 v = ((const float4*)(feat + src * DFEAT))[lane];
  float* ap = aggr + dst * DFEAT + (lane << 2);
  atomicAdd(ap + 0, v.x * w);
  atomicAdd(ap + 1, v.y * w);
  atomicAdd(ap + 2, v.z * w);
  atomicAdd(ap + 3, v.w * w);
}

// ---------------------------------------------------------------------------
// Fused GraphConv linear epilogue:
//   out = act( Arel @ Wrel + brel + Aroot @ Wroot )
// Arel/Aroot are [N_NODES,128] fp32 row-major; Wrel/Wroot are [128,NC].
// Grid: one block per 16-row M-tile (6250 blocks). NC/16 waves per block,
// one wave per 16-col N-tile. Uses V_WMMA_F32_16X16X4_F32, K-loop of 32
// steps, two WMMAs per step (rel + root paths into one accumulator).
//
// A-tiles staged in LDS (stride 132 floats to avoid 16-way bank conflicts),
// via GLOBAL_LOAD_ASYNC_TO_LDS_B128 (ASYNCcnt + s_wait_asynccnt + barrier).
// A fragment (16x4 fp32): lane m = lane%16; VGPR0/1 = K = 2*(lane>=16)+{0,1}.
// B fragment (4x16 fp32): lane n = lane%16; same K striping as A.
// C/D (16x16 fp32): VGPR j -> M = j + 8*(lane>=16), N = lane%16.
// ---------------------------------------------------------------------------
template <int NC, bool RELU>
__global__ __launch_bounds__(NC * 2) void gemm_kernel(
    const float* __restrict__ Arel, const float* __restrict__ Aroot,
    const float* __restrict__ Wrel, const float* __restrict__ brel,
    const float* __restrict__ Wroot, float* __restrict__ outp) {
  __shared__ float As_rel[16][132];
  __shared__ float As_root[16][132];

  const int r0 = blockIdx.x * 16;
  const int tid = threadIdx.x;
  constexpr int NTHREADS = NC * 2;  // (NC/16) waves * 32 lanes

  // Cooperative stage of both 16x128 A tiles into LDS, 16B per thread-step.
  // (132*4 = 528 bytes/row is 16B-aligned, so b128 transfers are legal.)
  for (int i = tid; i < 16 * 32; i += NTHREADS) {
    const int row = i >> 5;
    const int c4 = (i & 31) << 2;
    const size_t goff = (size_t)(r0 + row) * DFEAT + c4;
#if ASYNC_LDS
    __builtin_amdgcn_global_load_async_to_lds_b128(
        (__attribute__((address_space(1))) i4v*)(Arel + goff),
        (__attribute__((address_space(3))) i4v*)&As_rel[row][c4], 0, 0);
    __builtin_amdgcn_global_load_async_to_lds_b128(
        (__attribute__((address_space(1))) i4v*)(Aroot + goff),
        (__attribute__((address_space(3))) i4v*)&As_root[row][c4], 0, 0);
#else
    float4 va = *(const float4*)(Arel + goff);
    float4 vb = *(const float4*)(Aroot + goff);
    *(float4*)&As_rel[row][c4] = va;
    *(float4*)&As_root[row][c4] = vb;
#endif
  }
#if ASYNC_LDS
  wait_async_all();
#endif
  __syncthreads();

  const int wave = tid >> 5;
  const int lane = tid & 31;
  const int n = lane & 15;       // also A-fragment row m (both are lane%16)
  const int khalf = lane >> 4;   // selects K pair {0,1} vs {2,3}
  const int c0 = wave * 16;

  v8f acc = {};
#pragma unroll 4
  for (int k = 0; k < DFEAT; k += 4) {
    const int kk = k + 2 * khalf;
    // rel path
    v2f a = *(const v2f*)&As_rel[n][kk];
    v2f b;
    b.x = Wrel[kk * NC + c0 + n];
    b.y = Wrel[(kk + 1) * NC + c0 + n];
    acc = __builtin_amdgcn_wmma_f32_16x16x4_f32(false, a, false, b, (short)0,
                                                acc, false, false);
    // root path
    v2f a2 = *(const v2f*)&As_root[n][kk];
    v2f b2;
    b2.x = Wroot[kk * NC + c0 + n];
    b2.y = Wroot[(kk + 1) * NC + c0 + n];
    acc = __builtin_amdgcn_wmma_f32_16x16x4_f32(false, a2, false, b2, (short)0,
                                                acc, false, false);
  }

  const float bias = brel[c0 + n];
#pragma unroll
  for (int j = 0; j < 8; ++j) {
    float v = acc[j] + bias;
    if (RELU) v = fmaxf(v, 0.0f);
    const int m = j + 8 * khalf;
    outp[(size_t)(r0 + m) * NC + c0 + n] = v;
  }
}

// ---------------------------------------------------------------------------
// Orchestration:
//   zero(aggr); scatter(x)   -> aggr
//   gemm1: h = relu(aggr@Wrel1 + brel1 + x@Wroot1)
//   zero(aggr); scatter(h)   -> aggr
//   gemm2: out = aggr@Wrel2 + brel2 + h@Wroot2
// Workspace: aggr = N*128 floats (51.2MB, reused), h = N*128 floats (51.2MB).
// ---------------------------------------------------------------------------
extern "C" void kernel_launch(void* const* d_in, const int* in_sizes, int n_in,
                              void* d_out, int out_size, void* d_ws,
                              size_t ws_size, hipStream_t stream) {
  const float* x = (const float*)d_in[0];
  const float* ew = (const float*)d_in[1];
  const long long* eidx = (const long long*)d_in[2];  // int64 (2,E)
  const float* Wrel1 = (const float*)d_in[3];
  const float* brel1 = (const float*)d_in[4];
  const float* Wroot1 = (const float*)d_in[5];
  const float* Wrel2 = (const float*)d_in[6];
  const float* brel2 = (const float*)d_in[7];
  const float* Wroot2 = (const float*)d_in[8];
  float* out = (float*)d_out;

  const int E = in_sizes[1];  // edge_weight length

  float* aggr = (float*)d_ws;
  float* h = aggr + (size_t)N_NODES * DFEAT;

  const long long n4 = (long long)N_NODES * DFEAT / 4;
  const int zgrid = (int)((n4 + 255) / 256);
  const int sgrid = (E + 7) / 8;  // 8 edges (waves) per 256-thread block
  const int mgrid = N_NODES / 16; // 6250, exact

  // Layer 1
  zero_kernel<<<zgrid, 256, 0, stream>>>(aggr, n4);
  scatter_kernel<<<sgrid, 256, 0, stream>>>(x, eidx, ew, aggr, E);
  gemm_kernel<128, true><<<mgrid, 256, 0, stream>>>(aggr, x, Wrel1, brel1,
                                                    Wroot1, h);
  // Layer 2
  zero_kernel<<<zgrid, 256, 0, stream>>>(aggr, n4);
  scatter_kernel<<<sgrid, 256, 0, stream>>>(h, eidx, ew, aggr, E);
  gemm_kernel<64, false><<<mgrid, 128, 0, stream>>>(aggr, h, Wrel2, brel2,
                                                    Wroot2, out);
}